// BandSplit_19112604467686
// MI455X (gfx1250) — compile-verified
//
#include <hip/hip_runtime.h>

#define NB 34          // number of subbands
#define TT 1000        // time steps
#define FW 962         // 481*2 floats per (b,t) row
#define OC 128         // output channels
#define NT 63          // ceil(1000/16) time tiles

typedef float v2f __attribute__((ext_vector_type(2)));
typedef float v8f __attribute__((ext_vector_type(8)));

struct Tables {
    int C[NB];     // channels per band (2*sb)
    int KP[NB];    // C rounded up to multiple of 4
    int off[NB];   // 2*hz : float offset into the 962-wide row
    int wgoff[NB]; // float offset of padded Wg block in workspace
};
struct P34 { const float* p[NB]; };

// ---------------- Kernel 1: per-(band,batch) mean / invstd ----------------
__global__ void bs_stats(const float* __restrict__ x, float* __restrict__ mean,
                         float* __restrict__ inv, Tables tb) {
    const int i = blockIdx.x, b = blockIdx.y;
    const int C = tb.C[i], off = tb.off[i];
    const int N = C * TT;
    float s = 0.f, q = 0.f;
    for (int idx = threadIdx.x; idx < N; idx += 256) {
        int t = idx / C;
        int k = idx - t * C;
        float v = x[(b * TT + t) * FW + off + k];
        s += v; q += v * v;
    }
    __shared__ float rs[256], rq[256];
    rs[threadIdx.x] = s; rq[threadIdx.x] = q;
    __syncthreads();
    for (int w = 128; w > 0; w >>= 1) {
        if (threadIdx.x < w) {
            rs[threadIdx.x] += rs[threadIdx.x + w];
            rq[threadIdx.x] += rq[threadIdx.x + w];
        }
        __syncthreads();
    }
    if (threadIdx.x == 0) {
        float m  = rs[0] / (float)N;
        float va = rq[0] / (float)N - m * m;
        mean[i * 8 + b] = m;
        inv[i * 8 + b]  = rsqrtf(va + 1e-8f);
    }
}

// ---------------- Kernel 2: fold gamma into W, row sums ----------------
__global__ void bs_prep(Tables tb, P34 g, P34 be, P34 w,
                        float* __restrict__ Wg, float* __restrict__ Sg,
                        float* __restrict__ Sb) {
    const int i = blockIdx.x;
    const int C = tb.C[i], KP = tb.KP[i], wo = tb.wgoff[i];
    const float* W = w.p[i];
    const float* G = g.p[i];
    const float* Be = be.p[i];
    for (int idx = threadIdx.x; idx < OC * KP; idx += 256) {
        int row = idx / KP;
        int k   = idx - row * KP;
        float v = (k < C) ? W[row * C + k] * G[k] : 0.f;
        Wg[wo + idx] = v;
    }
    if (threadIdx.x < OC) {
        int o = threadIdx.x;
        float sg = 0.f, sb = 0.f;
        for (int k = 0; k < C; ++k) {
            float wv = W[o * C + k];
            sg += wv * G[k];
            sb += wv * Be[k];
        }
        Sg[i * OC + o] = sg;
        Sb[i * OC + o] = sb;
    }
}

// ---------------- Kernel 3: WMMA GEMM + fused norm affine ----------------
// grid = (band, ttile, batch); 256 threads = 8 waves; wave w -> output rows 16w..16w+15
__global__ void bs_gemm(const float* __restrict__ x, Tables tb, P34 fb,
                        const float* __restrict__ mean, const float* __restrict__ inv,
                        const float* __restrict__ Sg, const float* __restrict__ Sb,
                        const float* __restrict__ Wg, float* __restrict__ out) {
    const int i = blockIdx.x, tt = blockIdx.y, b = blockIdx.z;
    const int C = tb.C[i], KP = tb.KP[i], off = tb.off[i], wo = tb.wgoff[i];
    const int CS = KP + 2;              // LDS row stride (even -> 8B aligned, skewed banks)
    const int tbase = tt * 16;

    __shared__ float sX[16 * 122];      // max KP=120 -> CS=122

    // stage 16 time-columns of this band into LDS (zero-pad K and T tails)
    for (int idx = threadIdx.x; idx < 16 * KP; idx += 256) {
        int n = idx / KP;
        int k = idx - n * KP;
        int t = tbase + n;
        float v = 0.f;
        if (t < TT && k < C) v = x[(b * TT + t) * FW + off + k];
        sX[n * CS + k] = v;
    }
    __syncthreads();

    const int wave = threadIdx.x >> 5;
    const int lane = threadIdx.x & 31;
    const int half = lane >> 4;         // 0: K=k0,k0+1   1: K=k0+2,k0+3
    const int nl   = lane & 15;         // M index (A) / N index (B)

    const float* Arow = Wg + wo + (wave * 16 + nl) * KP + half * 2;  // global, L2-hot
    const float* Brow = sX + nl * CS + half * 2;                     // LDS

    v8f acc = {};
    for (int k = 0; k < KP; k += 4) {
        v2f a  = *(const v2f*)(Arow + k);
        v2f bb = *(const v2f*)(Brow + k);
        // D = A(16x4 f32) * B(4x16 f32) + C(16x16 f32)
        acc = __builtin_amdgcn_wmma_f32_16x16x4_f32(
            false, a, false, bb, (short)0, acc, false, false);
    }

    const float mu = mean[i * 8 + b];
    const float iv = inv[i * 8 + b];
    const float* fcb = fb.p[i];
    const int t = tbase + nl;           // N position of this lane's column

    #pragma unroll
    for (int r = 0; r < 8; ++r) {
        int o = wave * 16 + r + half * 8;
        float bias = Sb[i * OC + o] + fcb[o] - iv * mu * Sg[i * OC + o];
        if (t < TT)
            out[((b * OC + o) * TT + t) * NB + i] = iv * acc[r] + bias;
    }
}

extern "C" void kernel_launch(void* const* d_in, const int* in_sizes, int n_in,
                              void* d_out, int out_size, void* d_ws, size_t ws_size,
                              hipStream_t stream) {
    const float* x = (const float*)d_in[0];

    // band geometry (SUBBANDS = (5,) + (4,)*19 + (10,)*6 + (40,)*7 + (60,))
    int sb[NB]; int n = 0;
    sb[n++] = 5;
    for (int j = 0; j < 19; ++j) sb[n++] = 4;
    for (int j = 0; j < 6;  ++j) sb[n++] = 10;
    for (int j = 0; j < 7;  ++j) sb[n++] = 40;
    sb[n++] = 60;

    Tables tb;
    int hz = 0, wg = 0;
    for (int i = 0; i < NB; ++i) {
        int C = 2 * sb[i];
        tb.C[i]     = C;
        tb.KP[i]    = (C + 3) & ~3;
        tb.off[i]   = 2 * hz;
        tb.wgoff[i] = wg;
        wg += OC * tb.KP[i];
        hz += sb[i];
    }

    P34 g, be, w, fb;
    for (int i = 0; i < NB; ++i) {
        g.p[i]  = (const float*)d_in[1 + i];
        be.p[i] = (const float*)d_in[1 + NB + i];
        w.p[i]  = (const float*)d_in[1 + 2 * NB + i];
        fb.p[i] = (const float*)d_in[1 + 3 * NB + i];
    }

    // workspace layout (floats): mean[272] | inv[272] | Sg[4352] | Sb[4352] | Wg[...]
    float* ws   = (float*)d_ws;
    float* mean = ws;
    float* inv  = ws + NB * 8;
    float* Sg   = ws + 2 * NB * 8;
    float* Sb   = Sg + NB * OC;
    float* Wg   = Sb + NB * OC;

    bs_stats<<<dim3(NB, 8), 256, 0, stream>>>(x, mean, inv, tb);
    bs_prep<<<dim3(NB), 256, 0, stream>>>(tb, g, be, w, Wg, Sg, Sb);
    // band index fastest in grid -> neighboring-i output writes merge in the 192MB L2
    bs_gemm<<<dim3(NB, NT, 8), 256, 0, stream>>>(x, tb, fb, mean, inv, Sg, Sb, Wg,
                                                 (float*)d_out);
}